// GraphLearningModule_6262062317609
// MI455X (gfx1250) — compile-verified
//
#include <hip/hip_runtime.h>
#include <hip/hip_bf16.h>

// GAT (PyG GATConv x2 + LN + ELU + residual + mean-pool) for MI455X / gfx1250.
// Dense GEMMs run on V_WMMA_F32_16X16X4_F32 (fp32-exact). Edge phase is
// L2-resident scatter/gather with float atomics. Wave32 throughout.
//
// Workspace requirement: ~96 MB (x,h,y: 3*25.6MB; alpha: 13.6MB; per-node
// head scalars: ~3.2MB; pool + scalars: <64KB).

typedef float v2f __attribute__((ext_vector_type(2)));
typedef float v8f __attribute__((ext_vector_type(8)));

#define NEG_SLOPE 0.2f
#define LN_EPS 1e-5f

// ---------------------------------------------------------------------------
// GEMM: C[nrows x 128] = A[nrows x K] * B[K x 128] (+bias), fp32 WMMA 16x16x4.
// Block = 16-row strip (staged in LDS, padded), 8 waves x 16-col tiles.
// ---------------------------------------------------------------------------
template<int K>
__global__ __launch_bounds__(256)
void gemm_wmma_f32(const float* __restrict__ A, const float* __restrict__ B,
                   const float* __restrict__ bias, float* __restrict__ C,
                   int nrows)
{
    constexpr int KP = K + 4;            // pad: (K+4)%64==4 -> conflict-free
    __shared__ float As[16 * KP];
    const int m0  = blockIdx.x * 16;
    const int tid = threadIdx.x;

    for (int i = tid; i < 16 * K; i += 256) {   // uniform trip count
        int r = i / K, c = i - r * K;
        As[r * KP + c] = A[(size_t)(m0 + r) * K + c];
    }
    __syncthreads();

    const int lane = tid & 31;
    const int n0   = (tid >> 5) * 16;    // wave's column tile
    const int row  = lane & 15;
    const int kh   = (lane >> 4) * 2;    // lanes 0-15: K pair {0,1}; 16-31: {2,3}
    const int col  = n0 + row;

    v8f acc0 = {}; v8f acc1 = {};
    for (int k = 0; k < K; k += 8) {
        v2f a0, b0, a1, b1;
        a0.x = As[row * KP + k + kh];
        a0.y = As[row * KP + k + kh + 1];
        b0.x = B[(size_t)(k + kh) * 128 + col];
        b0.y = B[(size_t)(k + kh + 1) * 128 + col];
        a1.x = As[row * KP + k + 4 + kh];
        a1.y = As[row * KP + k + 4 + kh + 1];
        b1.x = B[(size_t)(k + 4 + kh) * 128 + col];
        b1.y = B[(size_t)(k + 4 + kh + 1) * 128 + col];
        acc0 = __builtin_amdgcn_wmma_f32_16x16x4_f32(false, a0, false, b0,
                                                     (short)0, acc0, false, false);
        acc1 = __builtin_amdgcn_wmma_f32_16x16x4_f32(false, a1, false, b1,
                                                     (short)0, acc1, false, false);
    }

    const float bv = bias ? bias[col] : 0.0f;
    const int   mb = m0 + ((lane >> 4) ? 8 : 0);  // C/D layout: vgpr j -> M=mb+j
#pragma unroll
    for (int j = 0; j < 8; ++j)
        C[(size_t)(mb + j) * 128 + col] = acc0[j] + acc1[j] + bv;
}

// ---------------------------------------------------------------------------
// Scalar prep: sum(edge_weight) -> mean; ce[l*4+h] = dot(We_l[h], ae_l[h]).
// ---------------------------------------------------------------------------
__global__ void reduce_sum_k(const float* __restrict__ w, float* acc, int E)
{
    int i = blockIdx.x * blockDim.x + threadIdx.x;
    float v = (i < E) ? w[i] : 0.0f;
    for (int off = 16; off; off >>= 1) v += __shfl_xor(v, off, 32);
    if ((threadIdx.x & 31) == 0) atomicAdd(acc, v);
}

__global__ void prep_k(const float* __restrict__ acc, float* mean_ew,
                       const float* __restrict__ We0, const float* __restrict__ ae0,
                       const float* __restrict__ We1, const float* __restrict__ ae1,
                       float* ce, int E)
{
    int t = threadIdx.x;
    if (t == 0) *mean_ew = *acc / (float)E;
    if (t < 8) {
        const float* We = (t < 4) ? We0 : We1;
        const float* ae = (t < 4) ? ae0 : ae1;
        int hh = t & 3;
        float s = 0.0f;
        for (int f = 0; f < 32; ++f) s += We[hh * 32 + f] * ae[hh * 32 + f];
        ce[t] = s;
    }
}

// ---------------------------------------------------------------------------
// Per-(node,head) attention scalars: a_src/a_dst = <h[n,head,:], att>.
// ---------------------------------------------------------------------------
__global__ void node_att_k(const float* __restrict__ h,
                           const float* __restrict__ as_, const float* __restrict__ ad_,
                           float* __restrict__ a_src, float* __restrict__ a_dst, int n)
{
    int i = blockIdx.x * blockDim.x + threadIdx.x;
    if (i >= n * 4) return;
    int node = i >> 2, hh = i & 3;
    const float* hv = h + (size_t)node * 128 + hh * 32;
    float ss = 0.0f, sd = 0.0f;
#pragma unroll 8
    for (int f = 0; f < 32; ++f) {
        float v = hv[f];
        ss += v * as_[hh * 32 + f];
        sd += v * ad_[hh * 32 + f];
    }
    a_src[i] = ss;
    a_dst[i] = sd;
}

// Order-preserving float->uint encoding for atomicMax.
__device__ __forceinline__ unsigned enc_f(float f) {
    unsigned b = __float_as_uint(f);
    return (b & 0x80000000u) ? ~b : (b | 0x80000000u);
}
__device__ __forceinline__ float dec_f(unsigned e) {
    unsigned b = (e & 0x80000000u) ? (e & 0x7FFFFFFFu) : ~e;
    return __uint_as_float(b);
}

// ---------------------------------------------------------------------------
// Edge pass 1: alpha = leakyrelu(a_src[s]+a_dst[d]+ew*ce); segment max (enc).
// ---------------------------------------------------------------------------
__global__ void edge_alpha_k(const int* __restrict__ ei, const float* __restrict__ ew,
                             const float* __restrict__ mean_ew,
                             const float* __restrict__ a_src, const float* __restrict__ a_dst,
                             const float* __restrict__ ce,
                             float* __restrict__ alpha, unsigned* __restrict__ amax_enc,
                             int E, int Et)
{
    int i = blockIdx.x * blockDim.x + threadIdx.x;
    if (i >= Et * 4) return;
    int e = i >> 2, hh = i & 3;
    int s, d; float w;
    if (e < E) { s = ei[e]; d = ei[E + e]; w = ew[e]; }
    else       { s = d = e - E;            w = *mean_ew; }
    float a = a_src[s * 4 + hh] + a_dst[d * 4 + hh] + w * ce[hh];
    a = (a > 0.0f) ? a : NEG_SLOPE * a;
    alpha[i] = a;
    atomicMax(&amax_enc[d * 4 + hh], enc_f(a));
}

// ---------------------------------------------------------------------------
// Edge pass 2: ex = exp(alpha - amax[d]); denom[d] += ex (ex stored in place).
// ---------------------------------------------------------------------------
__global__ void edge_exp_k(const int* __restrict__ ei, const unsigned* __restrict__ amax_enc,
                           float* __restrict__ alpha, float* __restrict__ denom,
                           int E, int Et)
{
    int i = blockIdx.x * blockDim.x + threadIdx.x;
    if (i >= Et * 4) return;
    int e = i >> 2, hh = i & 3;
    int d = (e < E) ? ei[E + e] : (e - E);
    float amax = dec_f(amax_enc[d * 4 + hh]);
    float ex = __expf(alpha[i] - amax);
    alpha[i] = ex;
    atomicAdd(&denom[d * 4 + hh], ex);
}

// ---------------------------------------------------------------------------
// Edge pass 3: one wave per edge; lane covers f = hh*32+lane over 4 heads.
// y[d,:] += h[s,:] * coef[head].
// ---------------------------------------------------------------------------
__global__ void edge_scatter_k(const int* __restrict__ ei, const float* __restrict__ hmat,
                               const float* __restrict__ ex, const float* __restrict__ denom,
                               float* __restrict__ y, int E, int Et)
{
    int wid = blockIdx.x * (blockDim.x >> 5) + (threadIdx.x >> 5);
    if (wid >= Et) return;
    int lane = threadIdx.x & 31;
    int s, d;
    if (wid < E) { s = ei[wid]; d = ei[E + wid]; }
    else         { s = d = wid - E; }
#pragma unroll
    for (int hh = 0; hh < 4; ++hh) {
        float coef = ex[(size_t)wid * 4 + hh] / (denom[d * 4 + hh] + 1e-16f);
        int f = hh * 32 + lane;
        atomicAdd(&y[(size_t)d * 128 + f], hmat[(size_t)s * 128 + f] * coef);
    }
}

// ---------------------------------------------------------------------------
// Post: x = elu(layernorm(y + b)) + x   (one wave per node, 4 feats/lane).
// ---------------------------------------------------------------------------
__global__ void post_ln_k(const float* __restrict__ y, const float* __restrict__ bias,
                          const float* __restrict__ g, const float* __restrict__ be,
                          float* __restrict__ x, int n)
{
    int node = blockIdx.x * (blockDim.x >> 5) + (threadIdx.x >> 5);
    if (node >= n) return;
    int lane = threadIdx.x & 31;
    float v[4]; float s = 0.0f;
#pragma unroll
    for (int k = 0; k < 4; ++k) {
        int f = lane + 32 * k;
        v[k] = y[(size_t)node * 128 + f] + bias[f];
        s += v[k];
    }
    for (int off = 16; off; off >>= 1) s += __shfl_xor(s, off, 32);
    float mu = s * (1.0f / 128.0f);
    float vs = 0.0f;
#pragma unroll
    for (int k = 0; k < 4; ++k) { float dd = v[k] - mu; vs += dd * dd; }
    for (int off = 16; off; off >>= 1) vs += __shfl_xor(vs, off, 32);
    float inv = rsqrtf(vs * (1.0f / 128.0f) + LN_EPS);
#pragma unroll
    for (int k = 0; k < 4; ++k) {
        int f = lane + 32 * k;
        float t = (v[k] - mu) * inv * g[f] + be[f];
        t = (t > 0.0f) ? t : expm1f(t);
        x[(size_t)node * 128 + f] = t + x[(size_t)node * 128 + f];
    }
}

// ---------------------------------------------------------------------------
// Mean pool over graphs.
// ---------------------------------------------------------------------------
__global__ void pool_acc_k(const float* __restrict__ x, const int* __restrict__ batch,
                           float* __restrict__ pool, float* __restrict__ cnt, int n)
{
    int node = blockIdx.x * (blockDim.x >> 5) + (threadIdx.x >> 5);
    if (node >= n) return;
    int lane = threadIdx.x & 31;
    int b = batch[node];
#pragma unroll
    for (int k = 0; k < 4; ++k) {
        int f = lane + 32 * k;
        atomicAdd(&pool[(size_t)b * 128 + f], x[(size_t)node * 128 + f]);
    }
    if (lane == 0) atomicAdd(&cnt[b], 1.0f);
}

__global__ void pool_final_k(const float* __restrict__ pool, const float* __restrict__ cnt,
                             float* __restrict__ out, int nB)
{
    int i = blockIdx.x * blockDim.x + threadIdx.x;
    if (i >= nB * 128) return;
    out[i] = pool[i] / fmaxf(cnt[i >> 7], 1.0f);
}

// ---------------------------------------------------------------------------
extern "C" void kernel_launch(void* const* d_in, const int* in_sizes, int n_in,
                              void* d_out, int out_size, void* d_ws, size_t ws_size,
                              hipStream_t stream)
{
    const float* node_feat = (const float*)d_in[0];
    const int*   ei        = (const int*)  d_in[1];   // [2,E] flat
    const float* ew        = (const float*)d_in[2];
    const int*   batch     = (const int*)  d_in[3];
    const float* Wp        = (const float*)d_in[4];
    const float* bp        = (const float*)d_in[5];

    const int N   = in_sizes[3];
    const int E   = in_sizes[2];
    const int Et  = E + N;                 // with self-loops
    const int B   = out_size / 128;

    // ---- workspace layout ----
    char* ws = (char*)d_ws;
    size_t off = 0;
    auto carve = [&](size_t bytes) -> char* {
        char* p = ws + off;
        off = (off + bytes + 255) & ~(size_t)255;
        return p;
    };
    float*    x      = (float*)   carve((size_t)N * 128 * 4);
    float*    h      = (float*)   carve((size_t)N * 128 * 4);
    float*    y      = (float*)   carve((size_t)N * 128 * 4);
    float*    alpha  = (float*)   carve((size_t)Et * 4 * 4);
    float*    a_src  = (float*)   carve((size_t)N * 4 * 4);
    float*    a_dst  = (float*)   carve((size_t)N * 4 * 4);
    unsigned* amax   = (unsigned*)carve((size_t)N * 4 * 4);
    float*    denom  = (float*)   carve((size_t)N * 4 * 4);
    float*    acc    = (float*)   carve(16);
    float*    meanew = (float*)   carve(16);
    float*    ce     = (float*)   carve(64);       // 8 floats (2 layers x 4 heads)
    float*    pool   = (float*)   carve((size_t)B * 128 * 4);
    float*    cnt    = (float*)   carve((size_t)B * 4);

    // ---- scalar prep: mean(edge_weight), ce per layer/head ----
    hipMemsetAsync(acc, 0, 4, stream);
    reduce_sum_k<<<(E + 255) / 256, 256, 0, stream>>>(ew, acc, E);
    prep_k<<<1, 32, 0, stream>>>(acc, meanew,
                                 (const float*)d_in[9],  (const float*)d_in[10],
                                 (const float*)d_in[17], (const float*)d_in[18],
                                 ce, E);

    // ---- input projection: x = node_feat @ Wp + bp (WMMA fp32) ----
    gemm_wmma_f32<384><<<N / 16, 256, 0, stream>>>(node_feat, Wp, bp, x, N);

    const int egrid   = (Et * 4 + 255) / 256;
    const int wgrid_e = (Et + 7) / 8;            // 8 waves per 256-thread block
    const int wgrid_n = (N + 7) / 8;

    for (int l = 0; l < 2; ++l) {
        const float* W    = (const float*)d_in[6 + l * 8];
        const float* as_  = (const float*)d_in[7 + l * 8];
        const float* ad_  = (const float*)d_in[8 + l * 8];
        const float* bGat = (const float*)d_in[11 + l * 8];
        const float* gLn  = (const float*)d_in[12 + l * 8];
        const float* beLn = (const float*)d_in[13 + l * 8];

        // h = x @ W (WMMA fp32)
        gemm_wmma_f32<128><<<N / 16, 256, 0, stream>>>(x, W, nullptr, h, N);

        node_att_k<<<(N * 4 + 255) / 256, 256, 0, stream>>>(h, as_, ad_, a_src, a_dst, N);

        hipMemsetAsync(amax,  0, (size_t)N * 4 * 4, stream);
        hipMemsetAsync(denom, 0, (size_t)N * 4 * 4, stream);
        hipMemsetAsync(y,     0, (size_t)N * 128 * 4, stream);

        edge_alpha_k<<<egrid, 256, 0, stream>>>(ei, ew, meanew, a_src, a_dst,
                                                ce + l * 4, alpha, amax, E, Et);
        edge_exp_k<<<egrid, 256, 0, stream>>>(ei, amax, alpha, denom, E, Et);
        edge_scatter_k<<<wgrid_e, 256, 0, stream>>>(ei, h, alpha, denom, y, E, Et);

        post_ln_k<<<wgrid_n, 256, 0, stream>>>(y, bGat, gLn, beLn, x, N);
    }

    // ---- global mean pool ----
    hipMemsetAsync(pool, 0, (size_t)B * 128 * 4, stream);
    hipMemsetAsync(cnt,  0, (size_t)B * 4, stream);
    pool_acc_k<<<wgrid_n, 256, 0, stream>>>(x, batch, pool, cnt, N);
    pool_final_k<<<(B * 128 + 255) / 256, 256, 0, stream>>>(pool, cnt, (float*)d_out, B);
}